// MapTRPostProcess_83021717832090
// MI455X (gfx1250) — compile-verified
//
#include <hip/hip_runtime.h>
#include <cstdint>

// MapTR NMS-free post-process for MI455X (gfx1250).
//
// Shapes (from reference): L=6, B=128, Q=2000, C=3, P=20, M(top-k)=100.
// Strategy: 1 workgroup per batch. Async-DMA the batch's 6000 logits straight
// into LDS (gfx1250 global_load_async_to_lds_b128, tracked by ASYNCcnt), do an
// exact radix-select of the 100th-largest key on monotonic uint keys fully in
// LDS, bitonic-sort the 128-padded winner set, then gather+decode only the
// 100 winning bbox/pts rows. Sigmoid applied only to the 100 winners.
// The pts gather (1000 independent float4 row-reads) is flattened across all
// 8 waves for maximum memory-level parallelism.

#define LAYERS  6
#define BATCH   128
#define NQ      2000
#define NC      3
#define NP      20
#define TOPK    100
#define NFLAT   (NQ * NC)       // 6000 scores per batch
#define NV4     (NFLAT / 4)     // 1500 16-byte async transfers per batch
#define NV4PTS  (NP * 2 / 4)    // 10 float4 per pts row
#define THREADS 256             // 8 wave32 per workgroup

// Output layout (floats): boxes | scores | labels | pts | mask
#define OFF_SCORES ((size_t)BATCH * TOPK * 4)                  //  51200
#define OFF_LABELS (OFF_SCORES + (size_t)BATCH * TOPK)         //  64000
#define OFF_PTS    (OFF_LABELS + (size_t)BATCH * TOPK)         //  76800
#define OFF_MASK   (OFF_PTS + (size_t)BATCH * TOPK * NP * 2)   // 588800

// ---- gfx1250 async tensor-path helpers (inline asm; CDNA5 ISA ch.10/15.18) ----
__device__ __forceinline__ void async_load_b128_to_lds(unsigned lds_byte_addr,
                                                       const void* gaddr) {
  // copies 16 bytes from gaddr into LDS at lds_byte_addr, per active lane; tracked by ASYNCcnt.
  asm volatile("global_load_async_to_lds_b128 %0, %1, off"
               :
               : "v"(lds_byte_addr), "v"(gaddr)
               : "memory");
}
__device__ __forceinline__ void wait_asynccnt0() {
  asm volatile("s_wait_asynccnt 0x0" ::: "memory");
}

__global__ __launch_bounds__(THREADS)
void maptr_postprocess_kernel(const float* __restrict__ cls,
                              const float* __restrict__ bbox,
                              const float* __restrict__ pts,
                              float* __restrict__ out) {
  __shared__ unsigned keys[NFLAT];     // logit bits -> ordered uint keys (24 KB)
  __shared__ unsigned hist[256];
  __shared__ unsigned sel_key[128];
  __shared__ unsigned sel_idx[128];
  __shared__ unsigned sh_bi[TOPK];     // winner -> bbox row index
  __shared__ unsigned sh_okm[TOPK];    // winner -> post-center-range mask
  __shared__ unsigned sh_prefix, sh_mask, sh_kk, sh_cnt_gt, sh_cnt_eq;

  const int tid = (int)threadIdx.x;
  const int b   = (int)blockIdx.x;

  // Last decoder layer only (as in MapTRNMSFreeCoder.decode).
  const float* cls_b  = cls  + ((size_t)(LAYERS - 1) * BATCH + b) * (size_t)NFLAT;
  const float* bbox_b = bbox + ((size_t)(LAYERS - 1) * BATCH + b) * (size_t)NQ * 4;
  const float* pts_b  = pts  + ((size_t)(LAYERS - 1) * BATCH + b) * (size_t)NQ * NP * 2;

  // ---------- Phase 1: async DMA logits -> LDS (no VGPR staging) ----------
  {
    const unsigned lds0 = (unsigned)(uintptr_t)&keys[0];
    for (int i = tid; i < NV4; i += THREADS) {
      const unsigned off = (unsigned)i * 16u;
      async_load_b128_to_lds(lds0 + off, (const char*)cls_b + off);
    }
    wait_asynccnt0();          // each wave waits for its own transfers
  }
  __syncthreads();             // all waves' LDS writes complete

  // ---------- Phase 2: float bits -> monotonic uint keys (in place) ----------
  for (int i = tid; i < NFLAT; i += THREADS) {
    const unsigned u = keys[i];
    keys[i] = (u & 0x80000000u) ? ~u : (u | 0x80000000u);
  }
  if (tid == 0) { sh_prefix = 0u; sh_mask = 0u; sh_kk = TOPK; }
  __syncthreads();

  // ---------- Phase 3: exact radix-select of the TOPK-th largest key ----------
  for (int shift = 24; shift >= 0; shift -= 8) {
    hist[tid] = 0u;
    __syncthreads();
    const unsigned pfx = sh_prefix, msk = sh_mask;
    for (int i = tid; i < NFLAT; i += THREADS) {
      const unsigned k = keys[i];
      if ((k & msk) == pfx) atomicAdd(&hist[(k >> shift) & 255u], 1u);
    }
    __syncthreads();
    if (tid == 0) {
      unsigned kk = sh_kk, cum = 0u;
      for (int bin = 255; bin >= 0; --bin) {
        const unsigned h = hist[bin];
        cum += h;
        if (cum >= kk) {
          sh_prefix = pfx | ((unsigned)bin << shift);
          sh_mask   = msk | (0xFFu << shift);
          sh_kk     = kk - (cum - h);   // eq-threshold entries still needed
          break;
        }
      }
    }
    __syncthreads();
  }

  const unsigned T        = sh_prefix;      // exact threshold key
  const unsigned needEq   = sh_kk;          // how many == T to take
  const unsigned cntAbove = TOPK - needEq;  // how many are strictly > T
  if (tid == 0) { sh_cnt_gt = 0u; sh_cnt_eq = 0u; }
  __syncthreads();

  // ---------- Phase 4: compact the TOPK winners into LDS ----------
  for (int i = tid; i < NFLAT; i += THREADS) {
    const unsigned k = keys[i];
    if (k > T) {
      const unsigned p = atomicAdd(&sh_cnt_gt, 1u);
      sel_key[p] = k;
      sel_idx[p] = (unsigned)i;
    } else if (k == T) {
      const unsigned p = atomicAdd(&sh_cnt_eq, 1u);
      if (p < needEq) {
        sel_key[cntAbove + p] = k;
        sel_idx[cntAbove + p] = (unsigned)i;
      }
    }
  }
  if (tid >= TOPK && tid < 128) { sel_key[tid] = 0u; sel_idx[tid] = 0u; }
  __syncthreads();

  // ---------- Phase 5: bitonic sort 128 entries, descending by key ----------
  for (unsigned size = 2; size <= 128; size <<= 1) {
    for (unsigned stride = size >> 1; stride > 0; stride >>= 1) {
      if (tid < 128) {
        const unsigned partner = (unsigned)tid ^ stride;
        if (partner > (unsigned)tid) {
          const unsigned ka = sel_key[tid],     ia = sel_idx[tid];
          const unsigned kb = sel_key[partner], ib = sel_idx[partner];
          const bool descBlock = (((unsigned)tid & size) == 0u);
          const bool doSwap = descBlock ? (ka < kb) : (ka > kb);
          if (doSwap) {
            sel_key[tid] = kb;     sel_idx[tid] = ib;
            sel_key[partner] = ka; sel_idx[partner] = ia;
          }
        }
      }
      __syncthreads();
    }
  }

  // ---------- Phase 6a: per-winner decode, mask, box/score/label/mask out ----
  // PC_RANGE = [-15,-30,-2, 15,30,2] -> sx=30, sy=60, x0=-15, y0=-30
  const float SX = 30.0f, SY = 60.0f, X0 = -15.0f, Y0 = -30.0f;
  if (tid < TOPK) {
    const unsigned key = sel_key[tid];
    const unsigned fi  = sel_idx[tid];
    // invert the monotonic transform to recover the logit, then sigmoid
    const unsigned ub = (key & 0x80000000u) ? (key & 0x7FFFFFFFu) : ~key;
    const float logit = __uint_as_float(ub);
    const float score = 1.0f / (1.0f + __expf(-logit));
    const unsigned label = fi % (unsigned)NC;  // flat idx = q*C + c
    const unsigned bi    = fi / (unsigned)NC;

    const float4 bb = *reinterpret_cast<const float4*>(bbox_b + (size_t)bi * 4);
    const float x1 = (bb.x - 0.5f * bb.z) * SX + X0;
    const float y1 = (bb.y - 0.5f * bb.w) * SY + Y0;
    const float x2 = (bb.x + 0.5f * bb.z) * SX + X0;
    const float y2 = (bb.y + 0.5f * bb.w) * SY + Y0;
    // POST_CENTER_RANGE = [-20,-35,-20,-35, 20,35,20,35]
    const bool ok = (x1 >= -20.f) && (x1 <= 20.f) && (y1 >= -35.f) && (y1 <= 35.f) &&
                    (x2 >= -20.f) && (x2 <= 20.f) && (y2 >= -35.f) && (y2 <= 35.f);

    sh_bi[tid]  = bi;
    sh_okm[tid] = ok ? 1u : 0u;

    const size_t om = (size_t)b * TOPK + (size_t)tid;
    reinterpret_cast<float4*>(out)[om] =
        ok ? make_float4(x1, y1, x2, y2) : make_float4(0.f, 0.f, 0.f, 0.f);
    out[OFF_SCORES + om] = ok ? score : 0.0f;
    out[OFF_LABELS + om] = ok ? (float)label : -1.0f;
    out[OFF_MASK + om]   = ok ? 1.0f : 0.0f;
  }
  __syncthreads();

  // ---------- Phase 6b: pts gather flattened over all 8 waves --------------
  // 100 winners x 10 float4 = 1000 independent 16B gathers; spread across all
  // 256 lanes so ~4 loads are in flight per lane (vs 10-deep chains on 100).
  for (int t = tid; t < TOPK * NV4PTS; t += THREADS) {
    const int m = t / NV4PTS;
    const int j = t - m * NV4PTS;
    const unsigned bi = sh_bi[m];
    const bool ok = (sh_okm[m] != 0u);

    const float4 v =
        reinterpret_cast<const float4*>(pts_b + (size_t)bi * NP * 2)[j];
    float4 r = make_float4(v.x * SX + X0, v.y * SY + Y0,
                           v.z * SX + X0, v.w * SY + Y0);
    if (!ok) r = make_float4(0.f, 0.f, 0.f, 0.f);

    float4* pd = reinterpret_cast<float4*>(out + OFF_PTS) +
                 ((size_t)b * TOPK + (size_t)m) * NV4PTS;
    pd[j] = r;
  }
}

extern "C" void kernel_launch(void* const* d_in, const int* in_sizes, int n_in,
                              void* d_out, int out_size, void* d_ws, size_t ws_size,
                              hipStream_t stream) {
  (void)in_sizes; (void)n_in; (void)d_ws; (void)ws_size; (void)out_size;
  const float* cls  = (const float*)d_in[0];  // [6,128,2000,3]
  const float* bbox = (const float*)d_in[1];  // [6,128,2000,4]
  const float* pts  = (const float*)d_in[2];  // [6,128,2000,20,2]
  maptr_postprocess_kernel<<<BATCH, THREADS, 0, stream>>>(cls, bbox, pts,
                                                          (float*)d_out);
}